// MyModule_21577915695469
// MI455X (gfx1250) — compile-verified
//
#include <hip/hip_runtime.h>
#include <hip/hip_bf16.h>

#define NUM_TASKS   8
#define NUM_TABLES  16
#define HASH        1000000
#define BATCH       8192
#define SEQ_L       20
#define NNZ         (BATCH * SEQ_L)

constexpr int kBlockThreads = 256;  // 8 waves (wave32)
constexpr int kWaves        = 8;
constexpr int kSegsPerWave  = 4;
constexpr int kSegsPerBlock = kWaves * kSegsPerWave;  // 32
constexpr int kChunk        = 2048;                   // ints staged in LDS (8KB)

#if defined(__gfx1250__) && __has_builtin(__builtin_amdgcn_tensor_load_to_lds)
#define HAVE_TDM 1
#else
#define HAVE_TDM 0
#endif

#if HAVE_TDM
typedef unsigned int v4u __attribute__((ext_vector_type(4)));
typedef int          v8i __attribute__((ext_vector_type(8)));
typedef int          v4i __attribute__((ext_vector_type(4)));
#endif

__global__ __launch_bounds__(kBlockThreads)
void embbag_pool_kernel(const float* __restrict__ weights,
                        const int*   __restrict__ offsets,
                        const int*   __restrict__ indices,
                        float*       __restrict__ out)
{
    __shared__ int smem[kChunk];

    const int t    = blockIdx.y;                       // table
    const int b0   = blockIdx.x * kSegsPerBlock;       // first segment of block
    const int lane = threadIdx.x & 31;
    const int wave = threadIdx.x >> 5;

    const int* off  = offsets + t * (BATCH + 1);
    const int* gidx = indices + (size_t)t * NNZ;
    const float* wbase = weights + (size_t)t * HASH;   // + k*(16*HASH) per task

    const int rs = off[b0];
    const int re = off[b0 + kSegsPerBlock];

    int segS[kSegsPerWave], segE[kSegsPerWave];
#pragma unroll
    for (int j = 0; j < kSegsPerWave; ++j) {
        const int b = b0 + wave * kSegsPerWave + j;
        segS[j] = off[b];
        segE[j] = off[b + 1];
    }

    float acc[kSegsPerWave][NUM_TASKS];
#pragma unroll
    for (int j = 0; j < kSegsPerWave; ++j)
#pragma unroll
        for (int k = 0; k < NUM_TASKS; ++k) acc[j][k] = 0.0f;

    for (int cs = rs; cs < re; cs += kChunk) {
        const int ce  = min(re, cs + kChunk);
        const int cnt = ce - cs;

#if HAVE_TDM
        if (wave == 0) {
            // ---- Tensor Data Mover: 1-D DMA of `cnt` dwords global -> LDS ----
            const unsigned long long ga = (unsigned long long)(uintptr_t)(gidx + cs);
            const unsigned int lds_off  = (unsigned int)(uintptr_t)&smem[0];
            const unsigned int c        = (unsigned int)cnt;

            v4u g0;
            g0.x = 1u;                                      // count=1 (valid), user mode
            g0.y = lds_off;                                 // lds_addr
            g0.z = (unsigned int)ga;                        // global_addr[31:0]
            g0.w = (unsigned int)(ga >> 32) | (2u << 30);   // global_addr[56:32] | type=2

            v8i g1;
            g1[0] = (int)0x00020000u;                           // data_size=2 (4B), no mcast
            g1[1] = (int)((c & 0xFFFFu) << 16);                 // tensor_dim0[15:0]
            g1[2] = (int)(((c >> 16) & 0xFFFFu) | (1u << 16));  // dim0 hi | tensor_dim1=1
            g1[3] = (int)((c & 0xFFFFu) << 16);                 // tile_dim0 = cnt
            g1[4] = 1;                                          // tile_dim1 = 1
            g1[5] = (int)c;                                     // tensor_dim0_stride
            g1[6] = 0;
            g1[7] = 0;

            v4i z4 = {0, 0, 0, 0};                              // <=2D: groups 2/3 unused
            v8i z8 = {0, 0, 0, 0, 0, 0, 0, 0};
            __builtin_amdgcn_tensor_load_to_lds(g0, g1, z4, z4, z8, 0);
#if __has_builtin(__builtin_amdgcn_s_wait_tensorcnt)
            __builtin_amdgcn_s_wait_tensorcnt(0);
#else
            asm volatile("s_wait_tensorcnt 0x0" ::: "memory");
#endif
        }
#else
        for (int j = threadIdx.x; j < cnt; j += kBlockThreads)
            smem[j] = gidx[cs + j];
#endif
        __syncthreads();

#pragma unroll
        for (int j = 0; j < kSegsPerWave; ++j) {
            const int lo = max(segS[j], cs);
            const int hi = min(segE[j], ce);
            for (int i = lo + lane; i < hi; i += 32) {
                const int idx = smem[i - cs];
                const float* wp = wbase + idx;
#pragma unroll
                for (int k = 0; k < NUM_TASKS; ++k)
                    acc[j][k] += wp[(size_t)k * ((size_t)NUM_TABLES * HASH)];
            }
        }
        __syncthreads();
    }

    // ---- wave32 butterfly reduction + scatter of pooled values ----
#pragma unroll
    for (int j = 0; j < kSegsPerWave; ++j) {
        const int b = b0 + wave * kSegsPerWave + j;
#pragma unroll
        for (int k = 0; k < NUM_TASKS; ++k) {
            float v = acc[j][k];
#pragma unroll
            for (int m = 16; m >= 1; m >>= 1)
                v += __shfl_xor(v, m, 32);
            if (lane == 0)
                out[((size_t)k * BATCH + b) * NUM_TABLES + t] = v;
        }
    }
}

extern "C" void kernel_launch(void* const* d_in, const int* in_sizes, int n_in,
                              void* d_out, int out_size, void* d_ws, size_t ws_size,
                              hipStream_t stream)
{
    const float* weights = (const float*)d_in[0];   // (8, 16, 1e6) f32
    const int*   offsets = (const int*)d_in[1];     // (16, 8193)  i32
    const int*   indices = (const int*)d_in[2];     // (16, 163840) i32
    float*       out     = (float*)d_out;           // (8, 8192, 16) f32

    dim3 grid(BATCH / kSegsPerBlock, NUM_TABLES);   // (256, 16)
    embbag_pool_kernel<<<grid, kBlockThreads, 0, stream>>>(weights, offsets, indices, out);
}